// LorentzAttention_1185410974345
// MI455X (gfx1250) — compile-verified
//
#include <hip/hip_runtime.h>
#include <math.h>

typedef float v2f __attribute__((ext_vector_type(2)));
typedef float v8f __attribute__((ext_vector_type(8)));
typedef int   i32v4 __attribute__((vector_size(16)));

#define QKD    68          // 65 (t+64 space) padded to multiple of 4
#define SCALE_ 0.125f      // 1/sqrt(64)
#define INVK_  10.0f       // 1/K
#define KCONST_ 0.1f
#define EPS_   1e-9f
#define SFAC_  2.8479424f  // exp(0.5*(digamma(256)-digamma(32)))

static __device__ __forceinline__ v8f wmma4(v2f a, v2f b, v8f c) {
  // D = A(16x4 f32) * B(4x16 f32) + C(16x16 f32)
  return __builtin_amdgcn_wmma_f32_16x16x4_f32(false, a, false, b, (short)0, c,
                                               false, false);
}

// --------------------------- CDNA5 async-to-LDS ----------------------------
#if __has_builtin(__builtin_amdgcn_global_load_async_to_lds_b32) && \
    __has_builtin(__builtin_amdgcn_global_load_async_to_lds_b128) && \
    __has_builtin(__builtin_amdgcn_s_wait_asynccnt)
#define HAVE_ASYNC 1
#else
#define HAVE_ASYNC 0
#endif

typedef __attribute__((address_space(1))) int*   gp32_t;
typedef __attribute__((address_space(3))) int*   lp32_t;
typedef __attribute__((address_space(1))) i32v4* gp128_t;
typedef __attribute__((address_space(3))) i32v4* lp128_t;

static __device__ __forceinline__ void async_copy_b32(const float* g, float* l) {
#if HAVE_ASYNC
  __builtin_amdgcn_global_load_async_to_lds_b32((gp32_t)g, (lp32_t)l, 0, 0);
#else
  *l = *g;
#endif
}
static __device__ __forceinline__ void async_copy_b128(const float* g, float* l) {
#if HAVE_ASYNC
  __builtin_amdgcn_global_load_async_to_lds_b128((gp128_t)g, (lp128_t)l, 0, 0);
#else
  *(float4*)l = *(const float4*)g;
#endif
}
static __device__ __forceinline__ void async_wait() {
#if HAVE_ASYNC
  __builtin_amdgcn_s_wait_asynccnt(0);
#endif
}

// ---------------------------------------------------------------------------
// Kernel A: QKV projection GEMM [8192 x 513] x Wsel^T (64 cols = 1 head) with
// Lorentz time epilogue. Writes q~=[qs,qt,0..], k~=[ks,-kt,0..], v~=[vt,vs,0..]
// each 68 floats wide, layout (B*H, N, 68).
// ---------------------------------------------------------------------------
__global__ __launch_bounds__(256) void qkv_kernel(
    const float* __restrict__ x,
    const float* __restrict__ Wq, const float* __restrict__ bq,
    const float* __restrict__ Wk, const float* __restrict__ bk,
    const float* __restrict__ Wv, const float* __restrict__ bv,
    float* __restrict__ Qh, float* __restrict__ Kh, float* __restrict__ Vh)
{
  __shared__ float Xl[128 * 68];   // 128 rows x 64 k, padded stride 68
  __shared__ float Wl[64 * 68];    // 64 cols x 64 k, padded stride 68
  const int rowbase = blockIdx.x * 128;
  const int grp  = blockIdx.y;          // 0..23
  const int which = grp >> 3;           // 0=q 1=k 2=v
  const int head  = grp & 7;
  const int tid = threadIdx.x;
  const int wave = tid >> 5;
  const int lane = tid & 31;
  const int l16 = lane & 15;
  const int hh  = lane >> 4;

  const float* Wsel = (which == 0) ? Wq : ((which == 1) ? Wk : Wv);
  const float* bsel = (which == 0) ? bq : ((which == 1) ? bk : bv);
  const float* Wbase = Wsel + (size_t)head * 64 * 513;
  float* Obase = (which == 0) ? Qh : ((which == 1) ? Kh : Vh);

  v8f acc[4];
#pragma unroll
  for (int i = 0; i < 4; ++i)
#pragma unroll
    for (int r = 0; r < 8; ++r) acc[i][r] = 0.0f;

  for (int chunk = 0; chunk < 9; ++chunk) {
    const int k0 = chunk * 64;
    if (chunk < 8) {   // fully in-range: async fire-and-forget staging
      for (int idx = tid; idx < 128 * 64; idx += 256) {
        int rr = idx >> 6, kc = idx & 63;
        async_copy_b32(&x[(size_t)(rowbase + rr) * 513 + k0 + kc],
                       &Xl[rr * 68 + kc]);
      }
      for (int idx = tid; idx < 64 * 64; idx += 256) {
        int cc = idx >> 6, kc = idx & 63;
        async_copy_b32(&Wbase[(size_t)cc * 513 + k0 + kc], &Wl[cc * 68 + kc]);
      }
      async_wait();
    } else {           // tail chunk: guarded + zero pad
      for (int idx = tid; idx < 128 * 64; idx += 256) {
        int rr = idx >> 6, kc = idx & 63;
        int gk = k0 + kc;
        Xl[rr * 68 + kc] = (gk < 513) ? x[(size_t)(rowbase + rr) * 513 + gk] : 0.0f;
      }
      for (int idx = tid; idx < 64 * 64; idx += 256) {
        int cc = idx >> 6, kc = idx & 63;
        int gk = k0 + kc;
        Wl[cc * 68 + kc] = (gk < 513) ? Wbase[(size_t)cc * 513 + gk] : 0.0f;
      }
    }
    __syncthreads();
    const float* Xw = Xl + wave * 16 * 68;
#pragma unroll 4
    for (int kk = 0; kk < 16; ++kk) {
      const int kd = kk * 4 + hh * 2;
      v2f a = *(const v2f*)(Xw + l16 * 68 + kd);
#pragma unroll
      for (int nt = 0; nt < 4; ++nt) {
        v2f b = *(const v2f*)(Wl + (nt * 16 + l16) * 68 + kd);
        acc[nt] = wmma4(a, b, acc[nt]);
      }
    }
    __syncthreads();
  }

  // bias add (per output column)
#pragma unroll
  for (int nt = 0; nt < 4; ++nt) {
    float bb = bsel[head * 64 + nt * 16 + l16];
#pragma unroll
    for (int r = 0; r < 8; ++r) acc[nt][r] += bb;
  }

  // per-row sum of squares -> Lorentz time; write padded outputs
#pragma unroll
  for (int r = 0; r < 8; ++r) {
    float s = 0.0f;
#pragma unroll
    for (int nt = 0; nt < 4; ++nt) { float v = acc[nt][r]; s += v * v; }
    s += __shfl_xor(s, 1, 32);
    s += __shfl_xor(s, 2, 32);
    s += __shfl_xor(s, 4, 32);
    s += __shfl_xor(s, 8, 32);
    float tval = sqrtf(INVK_ + s);
    const int gr = rowbase + wave * 16 + r + 8 * hh;
    const int bb2 = gr >> 11, nn = gr & 2047;
    float* orow = Obase + ((size_t)(bb2 * 8 + head) * 2048 + nn) * QKD;
    if (which == 2) {            // v~ = [t, space...]
#pragma unroll
      for (int nt = 0; nt < 4; ++nt) orow[1 + nt * 16 + l16] = acc[nt][r];
      if (l16 == 0) orow[0] = tval;
    } else {                     // q~/k~ = [space..., (+/-)t]
#pragma unroll
      for (int nt = 0; nt < 4; ++nt) orow[nt * 16 + l16] = acc[nt][r];
      if (l16 == 0) orow[64] = (which == 1) ? -tval : tval;
    }
    if (l16 >= 1 && l16 <= 3) orow[64 + l16] = 0.0f;   // pad 65..67
  }
}

// ---------------------------------------------------------------------------
// Kernel B: flash attention, double-buffered async K/V staging.
// S^T = k~ . q~^T (queries on the WMMA N axis), online softmax,
// PV via m^T = v^T . P^T. Epilogue: Minkowski normalization.
// Key block = 32; next block's staging overlaps current block's WMMAs.
// ---------------------------------------------------------------------------
__global__ __launch_bounds__(256) void attn_kernel(
    const float* __restrict__ Qh, const float* __restrict__ Kh,
    const float* __restrict__ Vh, float* __restrict__ Fused,
    float* __restrict__ Tbuf)
{
  __shared__ float kl[2][32 * QKD];   // [key][dim], 2 x 8704 B
  __shared__ float vT[2][80 * 36];    // [dim][key] pad 36, 2 x 11520 B
  const int qtile = blockIdx.x;    // 0..15
  const int bh    = blockIdx.y;    // 0..31  (b*8 + head)
  const int tid = threadIdx.x;
  const int wave = tid >> 5;
  const int lane = tid & 31;
  const int l16 = lane & 15;
  const int hh  = lane >> 4;
  const size_t base = (size_t)bh * 2048 * QKD;
  const int qrow = qtile * 128 + wave * 16;

  // zero-init both vT buffers once (rows 68..79 stay zero => padded dims)
  for (int idx = tid; idx < 2 * 80 * 36; idx += 256) (&vT[0][0])[idx] = 0.0f;

  // async staging of one 32-key block into buffer `buf`
  auto stage = [&](int kb, int buf) {
    const float* Kg = Kh + base + (size_t)kb * 32 * QKD;
    float* klb = kl[buf];
    for (int idx = tid; idx < (32 * QKD) / 4; idx += 256)
      async_copy_b128(Kg + idx * 4, klb + idx * 4);
    const float* Vg = Vh + base + (size_t)kb * 32 * QKD;
    float* vtb = vT[buf];
    for (int idx = tid; idx < 32 * QKD; idx += 256) {
      int key = idx & 31, d = idx >> 5;   // transpose scatter, conflict-free
      async_copy_b32(Vg + (size_t)key * QKD + d, vtb + d * 36 + key);
    }
  };

  // q~ B-fragments for this wave's 16 queries (reused across all key blocks)
  v2f qf[17];
#pragma unroll
  for (int kk = 0; kk < 17; ++kk)
    qf[kk] = *(const v2f*)(Qh + base + (size_t)(qrow + l16) * QKD + kk * 4 + hh * 2);

  v8f oacc[5];
#pragma unroll
  for (int i = 0; i < 5; ++i)
#pragma unroll
    for (int r = 0; r < 8; ++r) oacc[i][r] = 0.0f;
  float runmax = -3.0e38f;
  float runsum = 0.0f;

  stage(0, 0);
  async_wait();
  __syncthreads();

  for (int kb = 0; kb < 64; ++kb) {
    const int cur = kb & 1;
    if (kb + 1 < 64) stage(kb + 1, cur ^ 1);   // overlap with compute below

    // S^T: 2 m-tiles of 16 keys x 16 queries, K=68
    const float* klc = kl[cur];
    v8f sacc[2];
#pragma unroll
    for (int mt = 0; mt < 2; ++mt) {
      v8f s;
#pragma unroll
      for (int r = 0; r < 8; ++r) s[r] = 0.0f;
#pragma unroll
      for (int kk = 0; kk < 17; ++kk) {
        v2f a = *(const v2f*)(klc + (mt * 16 + l16) * QKD + kk * 4 + hh * 2);
        s = wmma4(a, qf[kk], s);
      }
      sacc[mt] = s;
    }

    // online softmax over keys (key index = 16*mt + r + 8*hh per lane)
    float nm = runmax;
#pragma unroll
    for (int mt = 0; mt < 2; ++mt)
#pragma unroll
      for (int r = 0; r < 8; ++r) {
        sacc[mt][r] *= SCALE_;
        nm = fmaxf(nm, sacc[mt][r]);
      }
    nm = fmaxf(nm, __shfl_xor(nm, 16, 32));
    float corr = __expf(runmax - nm);
    float lsum = 0.0f;
#pragma unroll
    for (int mt = 0; mt < 2; ++mt)
#pragma unroll
      for (int r = 0; r < 8; ++r) {
        float p = __expf(sacc[mt][r] - nm);
        sacc[mt][r] = p;
        lsum += p;
      }
    lsum += __shfl_xor(lsum, 16, 32);
    runsum = runsum * corr + lsum;
    runmax = nm;
#pragma unroll
    for (int i = 0; i < 5; ++i)
#pragma unroll
      for (int r = 0; r < 8; ++r) oacc[i][r] *= corr;

    // PV: m^T[d][q] += v^T-frag x P^T-frag, K = 32 keys
    const float* vtc = vT[cur];
#pragma unroll
    for (int kk = 0; kk < 8; ++kk) {
      const int mt = kk >> 2;
      const int c  = kk & 3;
      const int hs = c >> 1;       // source half holding these key rows
      const int rb = (c & 1) * 4;  // source accumulator reg base
      v2f bf;
#pragma unroll
      for (int r = 0; r < 2; ++r) {
        float t0 = sacc[mt][rb + r];       // value needed by dest half 0
        float t1 = sacc[mt][rb + 2 + r];   // value needed by dest half 1
        float bvv;
        if (hs == 0) {
          float sx = __shfl_xor(t1, 16, 32);
          bvv = hh ? sx : t0;
        } else {
          float sx = __shfl_xor(t0, 16, 32);
          bvv = hh ? t1 : sx;
        }
        bf[r] = bvv;
      }
#pragma unroll
      for (int dt = 0; dt < 5; ++dt) {
        v2f a = *(const v2f*)(vtc + (dt * 16 + l16) * 36 + kk * 4 + hh * 2);
        oacc[dt] = wmma4(a, bf, oacc[dt]);
      }
    }
    async_wait();      // next block's staging complete (this wave)
    __syncthreads();   // all waves done (reads of cur + writes of nxt)
  }

  // epilogue: normalize by softmax sum, then Minkowski normalization
  float inv = 1.0f / runsum;
  float tot = 0.0f;
#pragma unroll
  for (int dt = 0; dt < 5; ++dt)
#pragma unroll
    for (int r = 0; r < 8; ++r) {
      float m = oacc[dt][r] * inv;
      oacc[dt][r] = m;
      tot += m * m;
    }
  tot += __shfl_xor(tot, 16, 32);
  float mt0 = oacc[0][0];                 // d=0 (time) lives on half 0
  float mto = __shfl_xor(mt0, 16, 32);
  float m_t = hh ? mto : mt0;
  float ssum = tot - m_t * m_t;           // sum of space^2 (pads are zero)
  float isq = 1.0f / sqrtf(fmaxf(KCONST_ * (m_t * m_t - ssum), EPS_));

  const int b = bh >> 3, head = bh & 7;
  const int n = qtile * 128 + wave * 16 + l16;
  float* frow = Fused + (size_t)(b * 2048 + n) * 513;
#pragma unroll
  for (int dt = 0; dt < 5; ++dt)
#pragma unroll
    for (int r = 0; r < 8; ++r) {
      int d = dt * 16 + r + 8 * hh;
      if (d >= 1 && d <= 64) frow[head * 64 + d] = SFAC_ * oacc[dt][r] * isq;
    }
  if (hh == 0) Tbuf[(size_t)bh * 2048 + n] = oacc[0][0] * isq;
}

// ---------------------------------------------------------------------------
// Kernel C1: fused time column t'
// ---------------------------------------------------------------------------
__global__ void tprime_kernel(const float* __restrict__ Tbuf,
                              float* __restrict__ Fused) {
  int idx = blockIdx.x * blockDim.x + threadIdx.x;
  if (idx >= 8192) return;
  int b = idx >> 11, n = idx & 2047;
  float ts = 0.0f;
#pragma unroll
  for (int h = 0; h < 8; ++h) {
    float t = Tbuf[(size_t)(b * 8 + h) * 2048 + n];
    ts += t * t + INVK_;
  }
  Fused[(size_t)idx * 513] = sqrtf(fmaxf(INVK_ + SFAC_ * SFAC_ * ts, EPS_));
}

// ---------------------------------------------------------------------------
// Kernel C2: output GEMM [8192 x 513] x Wo^T -> out[:,1:513], plus
// deterministic per-(row, colgroup) partial sums of squares into Sq.
// ---------------------------------------------------------------------------
__global__ __launch_bounds__(256) void out_kernel(
    const float* __restrict__ Fused, const float* __restrict__ Wo,
    const float* __restrict__ bo, float* __restrict__ Out,
    float* __restrict__ Sq)
{
  __shared__ float Xl[128 * 68];
  __shared__ float Wl[64 * 68];
  const int rowbase = blockIdx.x * 128;
  const int cg = blockIdx.y;            // 0..7, 64 output cols each
  const int tid = threadIdx.x;
  const int wave = tid >> 5;
  const int lane = tid & 31;
  const int l16 = lane & 15;
  const int hh  = lane >> 4;

  v8f acc[4];
#pragma unroll
  for (int i = 0; i < 4; ++i)
#pragma unroll
    for (int r = 0; r < 8; ++r) acc[i][r] = 0.0f;

  for (int chunk = 0; chunk < 9; ++chunk) {
    const int k0 = chunk * 64;
    if (chunk < 8) {
      for (int idx = tid; idx < 128 * 64; idx += 256) {
        int rr = idx >> 6, kc = idx & 63;
        async_copy_b32(&Fused[(size_t)(rowbase + rr) * 513 + k0 + kc],
                       &Xl[rr * 68 + kc]);
      }
      for (int idx = tid; idx < 64 * 64; idx += 256) {
        int cc = idx >> 6, kc = idx & 63;
        async_copy_b32(&Wo[(size_t)(cg * 64 + cc) * 513 + k0 + kc],
                       &Wl[cc * 68 + kc]);
      }
      async_wait();
    } else {
      for (int idx = tid; idx < 128 * 64; idx += 256) {
        int rr = idx >> 6, kc = idx & 63;
        int gk = k0 + kc;
        Xl[rr * 68 + kc] = (gk < 513) ? Fused[(size_t)(rowbase + rr) * 513 + gk] : 0.0f;
      }
      for (int idx = tid; idx < 64 * 64; idx += 256) {
        int cc = idx >> 6, kc = idx & 63;
        int gk = k0 + kc;
        Wl[cc * 68 + kc] = (gk < 513) ? Wo[(size_t)(cg * 64 + cc) * 513 + gk] : 0.0f;
      }
    }
    __syncthreads();
    const float* Xw = Xl + wave * 16 * 68;
#pragma unroll 4
    for (int kk = 0; kk < 16; ++kk) {
      const int kd = kk * 4 + hh * 2;
      v2f a = *(const v2f*)(Xw + l16 * 68 + kd);
#pragma unroll
      for (int nt = 0; nt < 4; ++nt) {
        v2f b = *(const v2f*)(Wl + (nt * 16 + l16) * 68 + kd);
        acc[nt] = wmma4(a, b, acc[nt]);
      }
    }
    __syncthreads();
  }

#pragma unroll
  for (int nt = 0; nt < 4; ++nt) {
    float bb = bo[cg * 64 + nt * 16 + l16];
#pragma unroll
    for (int r = 0; r < 8; ++r) acc[nt][r] += bb;
  }
#pragma unroll
  for (int r = 0; r < 8; ++r) {
    float s = 0.0f;
#pragma unroll
    for (int nt = 0; nt < 4; ++nt) { float v = acc[nt][r]; s += v * v; }
    s += __shfl_xor(s, 1, 32);
    s += __shfl_xor(s, 2, 32);
    s += __shfl_xor(s, 4, 32);
    s += __shfl_xor(s, 8, 32);
    const int gr = rowbase + wave * 16 + r + 8 * hh;
    float* orow = Out + (size_t)gr * 513;
#pragma unroll
    for (int nt = 0; nt < 4; ++nt)
      orow[1 + cg * 64 + nt * 16 + l16] = acc[nt][r];
    if (l16 == 0) Sq[(size_t)gr * 8 + cg] = s;
  }
}

// ---------------------------------------------------------------------------
// Kernel C3: final time column of output
// ---------------------------------------------------------------------------
__global__ void tfinal_kernel(const float* __restrict__ Sq,
                              float* __restrict__ Out) {
  int idx = blockIdx.x * blockDim.x + threadIdx.x;
  if (idx >= 8192) return;
  float s = 0.0f;
#pragma unroll
  for (int g = 0; g < 8; ++g) s += Sq[(size_t)idx * 8 + g];
  Out[(size_t)idx * 513] = sqrtf(INVK_ + s);
}

// ---------------------------------------------------------------------------
extern "C" void kernel_launch(void* const* d_in, const int* in_sizes, int n_in,
                              void* d_out, int out_size, void* d_ws, size_t ws_size,
                              hipStream_t stream) {
  (void)in_sizes; (void)n_in; (void)out_size; (void)ws_size;
  const float* x  = (const float*)d_in[0];
  const float* Wq = (const float*)d_in[1];
  const float* bq = (const float*)d_in[2];
  const float* Wk = (const float*)d_in[3];
  const float* bk = (const float*)d_in[4];
  const float* Wv = (const float*)d_in[5];
  const float* bv = (const float*)d_in[6];
  const float* Wo = (const float*)d_in[7];
  const float* bo = (const float*)d_in[8];
  float* out = (float*)d_out;

  char* ws = (char*)d_ws;
  size_t off = 0;
  auto take = [&](size_t bytes) -> float* {
    float* p = (float*)(ws + off);
    off += (bytes + 255) & ~(size_t)255;
    return p;
  };
  const size_t qkvBytes = (size_t)32 * 2048 * QKD * sizeof(float);
  float* Qh    = take(qkvBytes);
  float* Kh    = take(qkvBytes);
  float* Vh    = take(qkvBytes);
  float* Fused = take((size_t)8192 * 513 * sizeof(float));
  float* Tbuf  = take((size_t)32 * 2048 * sizeof(float));
  float* Sq    = take((size_t)8192 * 8 * sizeof(float));

  qkv_kernel<<<dim3(64, 24), 256, 0, stream>>>(x, Wq, bq, Wk, bk, Wv, bv,
                                               Qh, Kh, Vh);
  attn_kernel<<<dim3(16, 32), 256, 0, stream>>>(Qh, Kh, Vh, Fused, Tbuf);
  tprime_kernel<<<32, 256, 0, stream>>>(Tbuf, Fused);
  out_kernel<<<dim3(64, 8), 256, 0, stream>>>(Fused, Wo, bo, out, Sq);
  tfinal_kernel<<<32, 256, 0, stream>>>(Sq, out);
}